// SLXLinear_47304769798215
// MI455X (gfx1250) — compile-verified
//
#include <hip/hip_runtime.h>
#include <hip/hip_bf16.h>

typedef float v2f __attribute__((ext_vector_type(2)));
typedef float v8f __attribute__((ext_vector_type(8)));

constexpr int IN    = 2048;
constexpr int OUT   = 2048;
constexpr int MROWS = 4 * 4096;   // 16384 rows of x
constexpr int RANK  = 204;        // low-rank K (divisible by 4)
constexpr int RPAD  = 208;        // RANK padded to 16 for tiling

#define WMMA_F32(a, b, c) \
  __builtin_amdgcn_wmma_f32_16x16x4_f32(false, (a), false, (b), (short)0, (c), false, false)

// ---------------------------------------------------------------------------
// K1: gate g[m] = sigmoid(dot(x[m,:], gate_w) + gate_b). One wave per row.
// ---------------------------------------------------------------------------
__global__ __launch_bounds__(256)
void gate_kernel(const float* __restrict__ x,
                 const float* __restrict__ gate_w,
                 const float* __restrict__ gate_b,
                 float* __restrict__ g) {
  const int wave = (blockIdx.x * blockDim.x + threadIdx.x) >> 5;
  const int lane = threadIdx.x & 31;
  if (wave >= MROWS) return;
  const float* row = x + (size_t)wave * IN;
  float s = 0.f;
  #pragma unroll 4
  for (int i = lane; i < IN; i += 32) s += row[i] * gate_w[i];
  #pragma unroll
  for (int off = 16; off > 0; off >>= 1) s += __shfl_xor(s, off, 32);
  if (lane == 0) g[wave] = 1.f / (1.f + __expf(-(s + gate_b[0])));
}

// ---------------------------------------------------------------------------
// K2: attn[o][i] = (W1 @ W2)[o][i] * mask[o][i].  One 16x16 tile per wave.
//   A = W1 (row-major, M=o, K), B = W2 (row-major, K, N=i).
// ---------------------------------------------------------------------------
__global__ __launch_bounds__(256)
void attn_kernel(const float* __restrict__ w1,
                 const float* __restrict__ w2,
                 const float* __restrict__ mask,
                 float* __restrict__ attn) {
  const int tile = blockIdx.x * (blockDim.x >> 5) + (threadIdx.x >> 5);
  const int lane = threadIdx.x & 31;
  const int o0 = (tile >> 7) << 4;   // 128 i-tiles per o-tile row
  const int i0 = (tile & 127) << 4;
  const int hi = lane >> 4;          // lane half selects K pair
  const int lo = lane & 15;
  const int ka = hi << 1;            // K offset {0,2} for this lane

  const float* arow = w1 + (size_t)(o0 + lo) * IN;      // A row M=lo
  v8f c = {};
  for (int k = 0; k < IN; k += 4) {
    v2f a = *(const v2f*)(arow + k + ka);               // K = k+ka, k+ka+1
    v2f b;
    b.x = w2[(size_t)(k + ka)     * IN + i0 + lo];      // B[k][n=lo]
    b.y = w2[(size_t)(k + ka + 1) * IN + i0 + lo];
    c = WMMA_F32(a, b, c);
  }
  #pragma unroll
  for (int v = 0; v < 8; ++v) {
    const int om = o0 + v + (hi << 3);                  // C row = v + 8*hi
    const size_t idx = (size_t)om * IN + i0 + lo;
    attn[idx] = c[v] * mask[idx];
  }
}

// ---------------------------------------------------------------------------
// K3: tmp[m][r] = sum_i x[m][i] * W2[r][i], r in [0,208).  (cols 204..207 are
// junk and never consumed.)  A = x row-major; B[k][n] = W2[n][k] -> contiguous
// float2 per lane along K.
// ---------------------------------------------------------------------------
__global__ __launch_bounds__(256)
void lr1_kernel(const float* __restrict__ x,
                const float* __restrict__ w2,
                float* __restrict__ tmp) {
  const int NT = RPAD / 16;                      // 13 n-tiles
  const int tile = blockIdx.x * (blockDim.x >> 5) + (threadIdx.x >> 5);
  if (tile >= (MROWS / 16) * NT) return;
  const int lane = threadIdx.x & 31;
  const int m0 = (tile / NT) << 4;
  const int r0 = (tile % NT) << 4;
  const int hi = lane >> 4, lo = lane & 15, ka = hi << 1;

  const float* arow = x  + (size_t)(m0 + lo) * IN;
  const float* brow = w2 + (size_t)(r0 + lo) * IN;     // row of W2 = column n
  v8f c = {};
  for (int k = 0; k < IN; k += 4) {
    v2f a = *(const v2f*)(arow + k + ka);
    v2f b = *(const v2f*)(brow + k + ka);
    c = WMMA_F32(a, b, c);
  }
  #pragma unroll
  for (int v = 0; v < 8; ++v)
    tmp[(size_t)(m0 + v + (hi << 3)) * RPAD + r0 + lo] = c[v];
}

// ---------------------------------------------------------------------------
// K4: out[m][o] = g[m] * (x @ attn^T)[m][o] + (1-g[m]) * (tmp @ W1[:,:204]^T)
//                 + bias[o]
// 32x32 output tile per wave: 2 A-loads + 2 B-loads feed 4 WMMAs per K-step
// (1 load per WMMA instead of 2), amortizing address math and issue slots.
// B operands are contiguous float2 per lane (attn[o][i] along i=K; W1[o][r]
// along r=K).
// ---------------------------------------------------------------------------
__global__ __launch_bounds__(256)
void final_kernel(const float* __restrict__ x,
                  const float* __restrict__ attn,
                  const float* __restrict__ tmp,
                  const float* __restrict__ w1,
                  const float* __restrict__ g,
                  const float* __restrict__ bias,
                  float* __restrict__ out) {
  const int tile = blockIdx.x * (blockDim.x >> 5) + (threadIdx.x >> 5);
  const int lane = threadIdx.x & 31;
  const int m0 = (tile >> 6) << 5;   // 512 m-tiles of 32 rows
  const int o0 = (tile & 63) << 5;   // 64  o-tiles of 32 cols
  const int hi = lane >> 4, lo = lane & 15, ka = hi << 1;

  // ---- Pass 1: sparse component, K = IN ----
  const float* xr0 = x    + (size_t)(m0 + lo) * IN;
  const float* xr1 = xr0  + (size_t)16 * IN;
  const float* br0 = attn + (size_t)(o0 + lo) * IN;    // B[k=i][n=o]=attn[o][i]
  const float* br1 = br0  + (size_t)16 * IN;
  v8f c00 = {}, c01 = {}, c10 = {}, c11 = {};
  #pragma unroll 2
  for (int k = 0; k < IN; k += 4) {
    v2f a0 = *(const v2f*)(xr0 + k + ka);
    v2f a1 = *(const v2f*)(xr1 + k + ka);
    v2f b0 = *(const v2f*)(br0 + k + ka);
    v2f b1 = *(const v2f*)(br1 + k + ka);
    c00 = WMMA_F32(a0, b0, c00);
    c01 = WMMA_F32(a0, b1, c01);
    c10 = WMMA_F32(a1, b0, c10);
    c11 = WMMA_F32(a1, b1, c11);
  }

  // ---- Pass 2: low-rank component, K = RANK (204, stops before pad) ----
  const float* tr0 = tmp + (size_t)(m0 + lo) * RPAD;
  const float* tr1 = tr0 + (size_t)16 * RPAD;
  const float* wr0 = w1  + (size_t)(o0 + lo) * IN;     // B[k=r][n=o]=W1[o][r]
  const float* wr1 = wr0 + (size_t)16 * IN;
  v8f d00 = {}, d01 = {}, d10 = {}, d11 = {};
  #pragma unroll 2
  for (int r = 0; r < RANK; r += 4) {
    v2f a0 = *(const v2f*)(tr0 + r + ka);
    v2f a1 = *(const v2f*)(tr1 + r + ka);
    v2f b0 = *(const v2f*)(wr0 + r + ka);
    v2f b1 = *(const v2f*)(wr1 + r + ka);
    d00 = WMMA_F32(a0, b0, d00);
    d01 = WMMA_F32(a0, b1, d01);
    d10 = WMMA_F32(a1, b0, d10);
    d11 = WMMA_F32(a1, b1, d11);
  }

  // ---- Epilogue: gated blend + bias ----
  const float bn0 = bias[o0 + lo];
  const float bn1 = bias[o0 + 16 + lo];
  #pragma unroll
  for (int v = 0; v < 8; ++v) {
    const int ma = m0 + v + (hi << 3);          // rows of sub-tile 0
    const int mb = ma + 16;                     // rows of sub-tile 1
    const float ga = g[ma], gb = g[mb];
    float* oa = out + (size_t)ma * OUT + o0 + lo;
    float* ob = out + (size_t)mb * OUT + o0 + lo;
    oa[0]  = ga * c00[v] + (1.f - ga) * d00[v] + bn0;
    oa[16] = ga * c01[v] + (1.f - ga) * d01[v] + bn1;
    ob[0]  = gb * c10[v] + (1.f - gb) * d10[v] + bn0;
    ob[16] = gb * c11[v] + (1.f - gb) * d11[v] + bn1;
  }
}

// ---------------------------------------------------------------------------
extern "C" void kernel_launch(void* const* d_in, const int* in_sizes, int n_in,
                              void* d_out, int out_size, void* d_ws, size_t ws_size,
                              hipStream_t stream) {
  const float* x      = (const float*)d_in[0];   // [16384, 2048]
  const float* w1     = (const float*)d_in[1];   // [2048, 2048]
  const float* w2     = (const float*)d_in[2];   // [2048, 2048]
  const float* gate_w = (const float*)d_in[3];   // [2048]
  const float* gate_b = (const float*)d_in[4];   // [1]
  const float* bias   = (const float*)d_in[5];   // [2048]
  const float* mask   = (const float*)d_in[6];   // [2048, 2048]
  float* out = (float*)d_out;

  // Workspace layout (floats): attn | tmp | g   (~29 MiB total)
  float* ws_attn = (float*)d_ws;
  float* ws_tmp  = ws_attn + (size_t)OUT * IN;
  float* ws_g    = ws_tmp  + (size_t)MROWS * RPAD;

  // K1: gate (16384 waves, 8 waves/block)
  gate_kernel<<<MROWS / 8, 256, 0, stream>>>(x, gate_w, gate_b, ws_g);

  // K2: masked attn (128*128 16x16 tiles, 8 tiles/block)
  attn_kernel<<<(128 * 128) / 8, 256, 0, stream>>>(w1, w2, mask, ws_attn);

  // K3: low-rank stage 1 (1024*13 16x16 tiles)
  {
    const int tiles = (MROWS / 16) * (RPAD / 16);
    lr1_kernel<<<(tiles + 7) / 8, 256, 0, stream>>>(x, w2, ws_tmp);
  }

  // K4: fused dual-GEMM + gated blend epilogue (512*64 32x32 tiles)
  {
    const int tiles = (MROWS / 32) * (OUT / 32);   // 32768
    final_kernel<<<tiles / 8, 256, 0, stream>>>(x, ws_attn, ws_tmp, w1,
                                                ws_g, bias, out);
  }
}